// YOLOLoss_27831388078489
// MI455X (gfx1250) — compile-verified
//
#include <hip/hip_runtime.h>

// ---------------- problem constants (from reference) ----------------
#define BS        32
#define NT        20
#define NUM_CLS   80
#define NANCH     3
#define IMG_WH    416.0f
#define EPSF      1e-7f
#define IGNORE    0.5f

#define W0 13
#define W1 26
#define W2 52
#define CNT0 (W0*W0*NANCH)              // 507
#define CNT1 (W1*W1*NANCH)              // 2028
#define OFF1 CNT0                       // 507
#define OFF2 (CNT0+CNT1)                // 2535
#define TOTAL (OFF2 + W2*W2*NANCH)      // 10647
#define NPOS  (BS*TOTAL)                // 340704
#define BLK   256
#define NBLK  ((NPOS + BLK - 1)/BLK)    // 1331

typedef __attribute__((ext_vector_type(2))) float v2f;
typedef __attribute__((ext_vector_type(8))) float v8f;

__device__ __forceinline__ float sigmoidf_(float x) { return 1.0f/(1.0f+expf(-x)); }

// IoU of two boxes in cx,cy,w,h form (matches _iou_cxcywh incl. EPS clamp)
__device__ __forceinline__ float iou_cxcywh(float ax, float ay, float aw, float ah,
                                            float bx, float by, float bw, float bh)
{
    float a_minx = ax - aw*0.5f, a_maxx = ax + aw*0.5f;
    float a_miny = ay - ah*0.5f, a_maxy = ay + ah*0.5f;
    float b_minx = bx - bw*0.5f, b_maxx = bx + bw*0.5f;
    float b_miny = by - bh*0.5f, b_maxy = by + bh*0.5f;
    float iw = fmaxf(fminf(a_maxx, b_maxx) - fmaxf(a_minx, b_minx), 0.0f);
    float ih = fmaxf(fminf(a_maxy, b_maxy) - fmaxf(a_miny, b_miny), 0.0f);
    float inter = iw*ih;
    float ua = aw*ah + bw*bh - inter;
    return inter / fmaxf(ua, EPSF);
}

__device__ __forceinline__ float ciou_(float px, float py, float pw, float ph,
                                       float tx, float ty, float tw, float th)
{
    float iou = iou_cxcywh(px, py, pw, ph, tx, ty, tw, th);
    float dx = px - tx, dy = py - ty;
    float center_d = dx*dx + dy*dy;
    float e_minx = fminf(px - pw*0.5f, tx - tw*0.5f);
    float e_maxx = fmaxf(px + pw*0.5f, tx + tw*0.5f);
    float e_miny = fminf(py - ph*0.5f, ty - th*0.5f);
    float e_maxy = fmaxf(py + ph*0.5f, ty + th*0.5f);
    float ex = e_maxx - e_minx, ey = e_maxy - e_miny;
    float diag = ex*ex + ey*ey;
    float ciou = iou - center_d / fmaxf(diag, EPSF);
    const float k = 4.0f / (3.14159265358979323846f*3.14159265358979323846f);
    float da = atanf(pw / fmaxf(ph, EPSF)) - atanf(tw / fmaxf(th, EPSF));
    float v = k * da * da;
    float alpha = v / fmaxf(1.0f - iou + v, EPSF);
    return ciou - alpha * v;
}

// -------- kernel 1: init obj_tid to -1 --------
__global__ void yolo_init_kernel(int* __restrict__ obj_tid, int n)
{
    int i = blockIdx.x*BLK + threadIdx.x;
    if (i < n) obj_tid[i] = -1;
}

// -------- kernel 2: target -> cell assignment (scatter) --------
__global__ void yolo_assign_kernel(const float* __restrict__ targets,
                                   const float* __restrict__ anchors,
                                   int* __restrict__ obj_tid)
{
    int i = blockIdx.x*BLK + threadIdx.x;       // i = b*NT + t
    if (i >= BS*NT) return;
    int b = i / NT;
    const float* tg = targets + (size_t)i*5;
    float cx = tg[0], cy = tg[1], tw = tg[2], th = tg[3];

    // argmax shape-IoU over 9 anchors (first max wins, like jnp.argmax)
    int   best  = 0;
    float bestv = -1.0f;
    for (int k = 0; k < 9; ++k) {
        float aw = anchors[2*k], ah = anchors[2*k+1];
        float inter = fminf(tw, aw) * fminf(th, ah);
        float siou  = inter / fmaxf(tw*th + aw*ah - inter, EPSF);
        if (siou > bestv) { bestv = siou; best = k; }
    }
    int level = best / NANCH;
    int a     = best - level*NANCH;
    int W     = (level == 0) ? W0 : (level == 1) ? W1 : W2;
    int off   = (level == 0) ? 0  : (level == 1) ? OFF1 : OFF2;
    float Wf  = (float)W;
    float Xf  = fminf(fmaxf(floorf(cx / IMG_WH * Wf), 0.0f), Wf - 1.0f);
    float Yf  = fminf(fmaxf(floorf(cy / IMG_WH * Wf), 0.0f), Wf - 1.0f);
    int X = (int)Xf, Y = (int)Yf;
    int flat = off + (Y*W + X)*NANCH + a;
    // sequential scatter semantics: later target overwrites; max(tidx) == last writer
    atomicMax(&obj_tid[b*TOTAL + flat], i);
}

// -------- kernel 3: per-site loss + WMMA wave reduction --------
__global__ void yolo_main_kernel(const float* __restrict__ f0,
                                 const float* __restrict__ f1,
                                 const float* __restrict__ f2,
                                 const float* __restrict__ targets,
                                 const float* __restrict__ anchors,
                                 const int*   __restrict__ obj_tid,
                                 float* __restrict__ partials)
{
    int idx = blockIdx.x*BLK + threadIdx.x;
    float partial = 0.0f;

    if (idx < NPOS) {
        int b = idx / TOTAL;
        int j = idx - b*TOTAL;

        const float* f; int W, off, lvl;
        if (j < OFF1)      { f = f0; W = W0; off = 0;    lvl = 0; }
        else if (j < OFF2) { f = f1; W = W1; off = OFF1; lvl = 1; }
        else               { f = f2; W = W2; off = OFF2; lvl = 2; }

        int local = j - off;
        int a     = local % NANCH;
        int cell  = local / NANCH;
        int X = cell % W, Y = cell / W;
        int HW = W*W;
        float sxy = IMG_WH / (float)W;

        const float* fp = f + ((size_t)b*(NANCH*(5+NUM_CLS)) + (size_t)a*(5+NUM_CLS))*HW
                            + (size_t)Y*W + X;
        float rx = fp[0];
        float ry = fp[(size_t)HW];
        float rw = fp[(size_t)2*HW];
        float rh = fp[(size_t)3*HW];
        float rc = fp[(size_t)4*HW];

        float aw = anchors[2*(lvl*NANCH + a)];
        float ah = anchors[2*(lvl*NANCH + a) + 1];

        float conf = sigmoidf_(rc);
        float p = fminf(fmaxf(conf, EPSF), 1.0f - EPSF);

        int tid = obj_tid[idx];
        bool obj = (tid >= 0);

        // ---- objectness / ignore mask ----
        if (obj) {
            partial += -logf(p);                    // bce * obj_label
        } else {
            float acx = ((float)X + 0.5f)*sxy;
            float acy = ((float)Y + 0.5f)*sxy;
            float maxiou = 0.0f;
            const float* tb = targets + (size_t)b*NT*5;
            for (int t = 0; t < NT; ++t) {
                float iou = iou_cxcywh(tb[5*t+0], tb[5*t+1], tb[5*t+2], tb[5*t+3],
                                       acx, acy, aw, ah);
                maxiou = fmaxf(maxiou, iou);
            }
            if (maxiou <= IGNORE)
                partial += -logf(1.0f - p);         // bce * noobj_mask
        }

        // ---- localization + classification at assigned sites ----
        if (obj) {
            const float* ts = targets + (size_t)tid*5;
            float tbx = ts[0], tby = ts[1], tbw = ts[2], tbh = ts[3];
            int label = (int)ts[4];
            float scale = 2.0f - tbw / IMG_WH * tbh / IMG_WH;

            float px = (sigmoidf_(rx) + (float)X)*sxy;
            float py = (sigmoidf_(ry) + (float)Y)*sxy;
            float pw = expf(rw)*aw;
            float ph = expf(rh)*ah;

            partial += (1.0f - ciou_(px, py, pw, ph, tbx, tby, tbw, tbh)) * scale;

            // log-softmax NLL over 80 class logits (strided by HW)
            const float* cp = fp + (size_t)5*HW;
            float m = -3.402823466e38f;
            for (int c = 0; c < NUM_CLS; ++c)
                m = fmaxf(m, cp[(size_t)c*HW]);
            float se = 0.0f;
            for (int c = 0; c < NUM_CLS; ++c)
                se += expf(cp[(size_t)c*HW] - m);
            float lse = m + logf(se);
            partial += -(cp[(size_t)label*HW] - lse);
        }
    }

    // ---- exact f32 wave-sum via V_WMMA_F32_16X16X4_F32 (B = ones) ----
    // A layout (32-bit, 16x4): VGPR0 holds K=0 (lanes 0-15, M=lane) and
    // K=2 (lanes 16-31, M=lane-16). With B = ones: D[m][n] = p_m + p_{m+16}.
    v2f A; A[0] = partial; A[1] = 0.0f;
    v2f Bv; Bv[0] = 1.0f;  Bv[1] = 1.0f;
    v8f C = {0.0f,0.0f,0.0f,0.0f,0.0f,0.0f,0.0f,0.0f};
    v8f D = __builtin_amdgcn_wmma_f32_16x16x4_f32(false, A, false, Bv,
                                                  (short)0, C, false, false);
    float s = D[0]+D[1]+D[2]+D[3]+D[4]+D[5]+D[6]+D[7];   // half-wave row sums
    s += __shfl_xor(s, 16, 32);                          // combine the two halves

    __shared__ float wsum[BLK/32];
    int lane = threadIdx.x & 31;
    int wv   = threadIdx.x >> 5;
    if (lane == 0) wsum[wv] = s;
    __syncthreads();
    if (threadIdx.x == 0) {
        float t = 0.0f;
        for (int i = 0; i < BLK/32; ++i) t += wsum[i];
        partials[blockIdx.x] = t;
    }
}

// -------- kernel 4: final reduction (double) and /bs --------
__global__ void yolo_reduce_kernel(const float* __restrict__ partials, int n,
                                   float* __restrict__ out)
{
    __shared__ double sd[BLK];
    double acc = 0.0;
    for (int i = threadIdx.x; i < n; i += BLK) acc += (double)partials[i];
    sd[threadIdx.x] = acc;
    __syncthreads();
    for (int st = BLK/2; st > 0; st >>= 1) {
        if (threadIdx.x < st) sd[threadIdx.x] += sd[threadIdx.x + st];
        __syncthreads();
    }
    if (threadIdx.x == 0) out[0] = (float)(sd[0] / (double)BS);
}

extern "C" void kernel_launch(void* const* d_in, const int* in_sizes, int n_in,
                              void* d_out, int out_size, void* d_ws, size_t ws_size,
                              hipStream_t stream)
{
    const float* f0      = (const float*)d_in[0];
    const float* f1      = (const float*)d_in[1];
    const float* f2      = (const float*)d_in[2];
    const float* targets = (const float*)d_in[3];
    const float* anchors = (const float*)d_in[4];

    // workspace layout
    int*   obj_tid  = (int*)d_ws;
    size_t obj_b    = ((size_t)NPOS*sizeof(int) + 255) & ~(size_t)255;
    float* partials = (float*)((char*)d_ws + obj_b);

    yolo_init_kernel<<<NBLK, BLK, 0, stream>>>(obj_tid, NPOS);
    yolo_assign_kernel<<<(BS*NT + BLK - 1)/BLK, BLK, 0, stream>>>(targets, anchors, obj_tid);
    yolo_main_kernel<<<NBLK, BLK, 0, stream>>>(f0, f1, f2, targets, anchors,
                                               obj_tid, partials);
    yolo_reduce_kernel<<<1, BLK, 0, stream>>>(partials, NBLK, (float*)d_out);
}